// MedGCN_50276887167361
// MI455X (gfx1250) — compile-verified
//
#include <hip/hip_runtime.h>

typedef float v2f __attribute__((ext_vector_type(2)));
typedef float v8f __attribute__((ext_vector_type(8)));

#define BM 128
#define BN 64
#define BK 16
#define PITCH 18          // even -> aligned b64 frag loads; 18*L mod 64 distinct for 16 lanes

enum { GF_ACC = 1, GF_RELU = 2, GF_TRANS = 4 };

// Generic fp32 WMMA GEMM:  C = [C +] op(A) (*Msk) @ B [+ bias] [relu]
//   op(A): MxK (row-major) or transpose of physical KxM
//   Msk  : optional elementwise multiplier, same layout as A (fused mask*adj)
//   B    : KxN row-major;  C : MxN row-major
// Block tile 128x64 (8 waves); each wave computes 32x32 via 4 f32 WMMA accumulators.
__global__ __launch_bounds__(256)
void gemm_wmma_f32(const float* __restrict__ A, const float* __restrict__ Msk,
                   const float* __restrict__ B, const float* __restrict__ bias,
                   float* __restrict__ C, int M, int N, int K, int flags)
{
    __shared__ __align__(16) float As[BM * PITCH];   // As[row][k]
    __shared__ __align__(16) float Bs[BN * PITCH];   // Bs[col][k]  (transposed-in-LDS)

    const int tid  = threadIdx.x;
    const int lane = tid & 31;
    const int wave = tid >> 5;
    const int wm   = wave & 3;      // m sub-tile: 32*wm
    const int wn   = wave >> 2;     // n sub-tile: 32*wn
    const int m0   = blockIdx.x * BM;
    const int n0   = blockIdx.y * BN;

    const int lhalf = lane >> 4;    // 0|1
    const int l15   = lane & 15;
    const int koff  = lhalf << 1;   // 0|2 : A/B fragment K split per ISA 7.12.2

    const bool trans   = (flags & GF_TRANS) != 0;
    const bool useMask = (Msk != nullptr);
    const int  lda     = trans ? M : K;

    v8f acc00 = {0.f,0.f,0.f,0.f,0.f,0.f,0.f,0.f};
    v8f acc01 = acc00, acc10 = acc00, acc11 = acc00;

    const int arow0 = (32 * wm + l15) * PITCH;
    const int arow1 = arow0 + 16 * PITCH;
    const int bcol0 = (32 * wn + l15) * PITCH;
    const int bcol1 = bcol0 + 16 * PITCH;

    const bool fullM = (m0 + BM) <= M;
    const bool fullN = (n0 + BN) <= N;

    for (int k0 = 0; k0 < K; k0 += BK) {
        const bool fullK = (k0 + BK) <= K;

        // ---- stage A tile (BM x BK), fusing mask and transpose ----
        if (fullM && fullK) {
            #pragma unroll
            for (int j = 0; j < 8; ++j) {
                int idx = tid + j * 256;                    // 0..2047
                int r, kk;
                if (trans) { kk = idx >> 7; r = idx & 127; }    // tid -> consecutive gm
                else       { r = idx >> 4; kk = idx & 15;  }    // tid -> consecutive gk
                long off = trans ? ((long)(k0 + kk) * lda + (m0 + r))
                                 : ((long)(m0 + r) * lda + (k0 + kk));
                float v = A[off];
                if (useMask) v *= Msk[off];
                As[r * PITCH + kk] = v;
            }
        } else {
            #pragma unroll
            for (int j = 0; j < 8; ++j) {
                int idx = tid + j * 256;
                int r, kk;
                if (trans) { kk = idx >> 7; r = idx & 127; }
                else       { r = idx >> 4; kk = idx & 15;  }
                int gm = m0 + r, gk = k0 + kk;
                float v = 0.f;
                if (gm < M && gk < K) {
                    long off = trans ? ((long)gk * lda + gm) : ((long)gm * lda + gk);
                    v = A[off];
                    if (useMask) v *= Msk[off];
                }
                As[r * PITCH + kk] = v;
            }
        }
        // ---- stage B tile (BK x BN) -> transposed layout Bs[col][k] ----
        if (fullN && fullK) {
            #pragma unroll
            for (int j = 0; j < 4; ++j) {
                int idx = tid + j * 256;                    // 0..1023
                int kk = idx >> 6, nn = idx & 63;           // tid -> consecutive gn (coalesced)
                Bs[nn * PITCH + kk] = B[(long)(k0 + kk) * N + (n0 + nn)];
            }
        } else {
            #pragma unroll
            for (int j = 0; j < 4; ++j) {
                int idx = tid + j * 256;
                int kk = idx >> 6, nn = idx & 63;
                int gk = k0 + kk, gn = n0 + nn;
                float v = 0.f;
                if (gk < K && gn < N) v = B[(long)gk * N + gn];
                Bs[nn * PITCH + kk] = v;
            }
        }

        // prefetch next k-tile into cache while we compute (global_prefetch_b8)
        if (k0 + BK < K) {
            long offA = trans ? ((long)(k0 + BK) * lda + m0 + (tid & 127))
                              : ((long)(m0 + (tid >> 1)) * lda + k0 + BK);
            __builtin_prefetch(&A[offA], 0, 1);
            __builtin_prefetch(&B[(long)(k0 + BK + (tid >> 6)) * N + n0 + (tid & 63)], 0, 1);
        }

        __syncthreads();

        #pragma unroll
        for (int kk = 0; kk < BK; kk += 4) {
            // aligned contiguous b64 fragment loads (index even: kk%4==0, koff in {0,2})
            v2f a0 = *(const v2f*)&As[arow0 + kk + koff];
            v2f a1 = *(const v2f*)&As[arow1 + kk + koff];
            v2f b0 = *(const v2f*)&Bs[bcol0 + kk + koff];
            v2f b1 = *(const v2f*)&Bs[bcol1 + kk + koff];
            acc00 = __builtin_amdgcn_wmma_f32_16x16x4_f32(false, a0, false, b0,
                                                          (short)0, acc00, false, false);
            acc01 = __builtin_amdgcn_wmma_f32_16x16x4_f32(false, a0, false, b1,
                                                          (short)0, acc01, false, false);
            acc10 = __builtin_amdgcn_wmma_f32_16x16x4_f32(false, a1, false, b0,
                                                          (short)0, acc10, false, false);
            acc11 = __builtin_amdgcn_wmma_f32_16x16x4_f32(false, a1, false, b1,
                                                          (short)0, acc11, false, false);
        }
        __syncthreads();
    }

    // ---- epilogue: C/D layout = 8 VGPRs, M=r (lanes<16) / r+8 (lanes>=16) ----
    const int rb0 = m0 + 32 * wm + (lhalf << 3);
    const int rb1 = rb0 + 16;
    const int c0  = n0 + 32 * wn + l15;
    const int c1  = c0 + 16;

    #pragma unroll
    for (int r = 0; r < 8; ++r) {
        int row0 = rb0 + r, row1 = rb1 + r;
        float v;
        if (row0 < M && c0 < N) {
            v = acc00[r];
            if (bias)            v += bias[c0];
            if (flags & GF_ACC)  v += C[(long)row0 * N + c0];
            if (flags & GF_RELU) v = fmaxf(v, 0.f);
            C[(long)row0 * N + c0] = v;
        }
        if (row0 < M && c1 < N) {
            v = acc01[r];
            if (bias)            v += bias[c1];
            if (flags & GF_ACC)  v += C[(long)row0 * N + c1];
            if (flags & GF_RELU) v = fmaxf(v, 0.f);
            C[(long)row0 * N + c1] = v;
        }
        if (row1 < M && c0 < N) {
            v = acc10[r];
            if (bias)            v += bias[c0];
            if (flags & GF_ACC)  v += C[(long)row1 * N + c0];
            if (flags & GF_RELU) v = fmaxf(v, 0.f);
            C[(long)row1 * N + c0] = v;
        }
        if (row1 < M && c1 < N) {
            v = acc11[r];
            if (bias)            v += bias[c1];
            if (flags & GF_ACC)  v += C[(long)row1 * N + c1];
            if (flags & GF_RELU) v = fmaxf(v, 0.f);
            C[(long)row1 * N + c1] = v;
        }
    }
}

__global__ void ew_add_bias(float* __restrict__ out, const float* __restrict__ in,
                            const float* __restrict__ bias, long total, int cols)
{
    long i = (long)blockIdx.x * 256 + threadIdx.x;
    if (i < total) out[i] = in[i] + bias[i % cols];
}

__global__ void ew_relu(float* __restrict__ out, const float* __restrict__ in, long n)
{
    long i = (long)blockIdx.x * 256 + threadIdx.x;
    if (i < n) out[i] = fmaxf(in[i], 0.f);
}

// bsum0 = b0+b1+b2+b3 ; bsum1..3 = b0+b{1,2,3} ; bz0 = 4*bs ; bz1 = 2*bs
__global__ void bias_combine(const float* b0, const float* b1, const float* b2,
                             const float* b3, const float* bs,
                             float* s0, float* s1, float* s2, float* s3,
                             float* z0, float* z1)
{
    int c = threadIdx.x;
    if (c < 256) {
        float v0 = b0[c], v1 = b1[c], v2 = b2[c], v3 = b3[c];
        s0[c] = v0 + v1 + v2 + v3;
        s1[c] = v0 + v1;
        s2[c] = v0 + v2;
        s3[c] = v0 + v3;
    }
    if (c < 128) {
        float v = bs[c];
        z0[c] = 4.f * v;
        z1[c] = 2.f * v;
    }
}

static inline void gemm(const float* A, const float* Msk, const float* B,
                        const float* bias, float* C, int M, int N, int K,
                        int flags, hipStream_t s)
{
    dim3 grid((M + BM - 1) / BM, (N + BN - 1) / BN);
    gemm_wmma_f32<<<grid, 256, 0, s>>>(A, Msk, B, bias, C, M, N, K, flags);
}

static inline void addb(float* out, const float* in, const float* bias,
                        long M, int C, hipStream_t s)
{
    long tot = M * (long)C;
    ew_add_bias<<<(unsigned)((tot + 255) / 256), 256, 0, s>>>(out, in, bias, tot, C);
}

extern "C" void kernel_launch(void* const* d_in, const int* in_sizes, int n_in,
                              void* d_out, int out_size, void* d_ws, size_t ws_size,
                              hipStream_t stream)
{
    // ---- inputs (setup_inputs dict order) ----
    const float* fea0 = (const float*)d_in[0];   // [8000,1024]
    const float* fea1 = (const float*)d_in[1];   // [6000, 512]
    const float* fea2 = (const float*)d_in[2];   // [4000, 256]
    const float* fea3 = (const float*)d_in[3];   // [ 300, 300]
    const float* adj1 = (const float*)d_in[4];   // [8000,6000]
    const float* msk1 = (const float*)d_in[5];
    const float* adj2 = (const float*)d_in[6];   // [8000,4000]
    const float* msk2 = (const float*)d_in[7];
    const float* adj3 = (const float*)d_in[8];   // [8000, 300]
    const float* msk3 = (const float*)d_in[9];
    const float* W0 = (const float*)d_in[10]; const float* b0 = (const float*)d_in[11];
    const float* W1 = (const float*)d_in[12]; const float* b1 = (const float*)d_in[13];
    const float* W2 = (const float*)d_in[14]; const float* b2 = (const float*)d_in[15];
    const float* W3 = (const float*)d_in[16]; const float* b3 = (const float*)d_in[17];
    const float* Ws = (const float*)d_in[18]; const float* bs = (const float*)d_in[19];
    const float* Wp = (const float*)d_in[20]; const float* bp = (const float*)d_in[21];

    const int N0 = 8000, N1 = 6000, N2 = 4000, N3 = 300;
    const int D0 = 1024, D1 = 512, D2 = 256, D3 = 300;
    const int H0 = 256, H1 = 128;

    // ---- workspace layout (floats), ~51 MB ----
    float* w = (float*)d_ws;
    float* s0 = w;                       // 8000*256
    float* s1 = s0 + (long)N0 * H0;      // 6000*256
    float* s2 = s1 + (long)N1 * H0;      // 4000*256
    float* s3 = s2 + (long)N2 * H0;      //  300*256
    float* h0 = s3 + (long)N3 * H0;
    float* h1 = h0 + (long)N0 * H0;
    float* h2 = h1 + (long)N1 * H0;
    float* h3 = h2 + (long)N2 * H0;
    float* t0 = h3 + (long)N3 * H0;      // 8000*128
    float* t1 = t0 + (long)N0 * H1;
    float* t2 = t1 + (long)N1 * H1;
    float* t3 = t2 + (long)N2 * H1;
    float* r0 = t3 + (long)N3 * H1;      // relu(z0), 8000*128
    float* bsum0 = r0 + (long)N0 * H1;   // 256 each
    float* bsum1 = bsum0 + 256;
    float* bsum2 = bsum1 + 256;
    float* bsum3 = bsum2 + 256;
    float* bz0   = bsum3 + 256;          // 128
    float* bz1   = bz0 + 128;            // 128

    // ---- outputs: recon | z0 | z1 | z2 | z3 (flat concat) ----
    float* recon = (float*)d_out;                 // 8000*300
    float* z0 = recon + (long)N0 * D3;            // 8000*128
    float* z1 = z0 + (long)N0 * H1;               // 6000*128
    float* z2 = z1 + (long)N1 * H1;               // 4000*128
    float* z3 = z2 + (long)N2 * H1;               //  300*128

    bias_combine<<<1, 256, 0, stream>>>(b0, b1, b2, b3, bs,
                                        bsum0, bsum1, bsum2, bsum3, bz0, bz1);

    // ---- layer 0 encoders: s_i = fea_i @ W_i ----
    gemm(fea0, nullptr, W0, nullptr, s0, N0, H0, D0, 0, stream);
    gemm(fea1, nullptr, W1, nullptr, s1, N1, H0, D1, 0, stream);
    gemm(fea2, nullptr, W2, nullptr, s2, N2, H0, D2, 0, stream);
    gemm(fea3, nullptr, W3, nullptr, s3, N3, H0, D3, 0, stream);

    // ---- h0 = relu(s0 + Σb + A1@s1 + A2@s2 + A3@s3) ----
    addb(h0, s0, bsum0, N0, H0, stream);
    gemm(adj1, msk1, s1, nullptr, h0, N0, H0, N1, GF_ACC, stream);
    gemm(adj2, msk2, s2, nullptr, h0, N0, H0, N2, GF_ACC, stream);
    gemm(adj3, msk3, s3, nullptr, h0, N0, H0, N3, GF_ACC | GF_RELU, stream);

    // ---- h_j = relu(s_j + b0+b_j + A_j^T @ s0) ----
    addb(h1, s1, bsum1, N1, H0, stream);
    gemm(adj1, msk1, s0, nullptr, h1, N1, H0, N0, GF_TRANS | GF_ACC | GF_RELU, stream);
    addb(h2, s2, bsum2, N2, H0, stream);
    gemm(adj2, msk2, s0, nullptr, h2, N2, H0, N0, GF_TRANS | GF_ACC | GF_RELU, stream);
    addb(h3, s3, bsum3, N3, H0, stream);
    gemm(adj3, msk3, s0, nullptr, h3, N3, H0, N0, GF_TRANS | GF_ACC | GF_RELU, stream);

    // ---- layer 1 shared: t_i = h_i @ Ws ----
    gemm(h0, nullptr, Ws, nullptr, t0, N0, H1, H0, 0, stream);
    gemm(h1, nullptr, Ws, nullptr, t1, N1, H1, H0, 0, stream);
    gemm(h2, nullptr, Ws, nullptr, t2, N2, H1, H0, 0, stream);
    gemm(h3, nullptr, Ws, nullptr, t3, N3, H1, H0, 0, stream);

    // ---- z0 = t0 + 4*bs + A1@t1 + A2@t2 + A3@t3 (output, no relu) ----
    addb(z0, t0, bz0, N0, H1, stream);
    gemm(adj1, msk1, t1, nullptr, z0, N0, H1, N1, GF_ACC, stream);
    gemm(adj2, msk2, t2, nullptr, z0, N0, H1, N2, GF_ACC, stream);
    gemm(adj3, msk3, t3, nullptr, z0, N0, H1, N3, GF_ACC, stream);

    // ---- z_j = t_j + 2*bs + A_j^T @ t0 ----
    addb(z1, t1, bz1, N1, H1, stream);
    gemm(adj1, msk1, t0, nullptr, z1, N1, H1, N0, GF_TRANS | GF_ACC, stream);
    addb(z2, t2, bz1, N2, H1, stream);
    gemm(adj2, msk2, t0, nullptr, z2, N2, H1, N0, GF_TRANS | GF_ACC, stream);
    addb(z3, t3, bz1, N3, H1, stream);
    gemm(adj3, msk3, t0, nullptr, z3, N3, H1, N0, GF_TRANS | GF_ACC, stream);

    // ---- recon = relu(z0) @ Wp + bp ----
    {
        long n = (long)N0 * H1;
        ew_relu<<<(unsigned)((n + 255) / 256), 256, 0, stream>>>(r0, z0, n);
    }
    gemm(r0, nullptr, Wp, bp, recon, N0, D3, H1, 0, stream);
}